// GraphAttentionLayer_89799176225427
// MI455X (gfx1250) — compile-verified
//
#include <hip/hip_runtime.h>
#include <math.h>

// Problem constants (from reference): B=8, N=2048, IN=OUT=256, H=4, DH=64
#define BATCH   8
#define SEQ     2048
#define IN_DIM  256
#define OUT_DIM 256
#define NHEAD   4
#define DHEAD   64
#define ROWS    (BATCH * SEQ)   // 16384

typedef __attribute__((ext_vector_type(2))) float v2f;
typedef __attribute__((ext_vector_type(8))) float v8f;

// ---------------------------------------------------------------------------
// Kernel 1: fold attention vectors into the weight matrices.
//   u_s[k][h] = sum_d W_s[k, h*64+d] * attn[h, d]        (left projector)
//   u_n[k][h] = sum_d W_n[k, h*64+d] * attn[h, 64+d]     (right projector)
// This removes the entire Hs GEMM from the problem.
// ---------------------------------------------------------------------------
__global__ __launch_bounds__(256) void gat_prep_kernel(
    const float* __restrict__ W_s, const float* __restrict__ W_n,
    const float* __restrict__ attn, float* __restrict__ u_s,
    float* __restrict__ u_n) {
  const int k = threadIdx.x;  // 0..255
  #pragma unroll
  for (int hh = 0; hh < NHEAD; ++hh) {
    float ss = 0.f, sn = 0.f;
    for (int d = 0; d < DHEAD; ++d) {
      ss += W_s[k * OUT_DIM + hh * DHEAD + d] * attn[hh * 2 * DHEAD + d];
      sn += W_n[k * OUT_DIM + hh * DHEAD + d] * attn[hh * 2 * DHEAD + DHEAD + d];
    }
    u_s[k * NHEAD + hh] = ss;
    u_n[k * NHEAD + hh] = sn;
  }
}

// ---------------------------------------------------------------------------
// Kernel 2: Hn = h @ W_n  (16384x256 @ 256x256) with V_WMMA_F32_16X16X4_F32.
// One wave -> 16x64 output tile (4 accumulators), 4 waves/block cover 16x256.
// A (16x4 f32): lane<16 holds M=lane, K = {0,1}; lane>=16 holds M=lane-16,
// K = {2,3}  (per ISA 32-bit A layout). B (4x16) is the mirror with N in lanes.
// ---------------------------------------------------------------------------
__global__ __launch_bounds__(128) void gat_proj_kernel(
    const float* __restrict__ h, const float* __restrict__ Wn,
    float* __restrict__ Hn) {
  const int lane = threadIdx.x & 31;
  const int wave = threadIdx.x >> 5;   // 0..3 -> 64-wide column slab
  const int half = lane >> 4;          // which 16-lane half
  const int lm   = lane & 15;
  const int koff = half * 2;           // K offset for this half: 0 or 2
  const int row0 = blockIdx.x * 16;
  const int col0 = wave * 64;

  v8f acc[4];
  #pragma unroll
  for (int dt = 0; dt < 4; ++dt)
    #pragma unroll
    for (int v = 0; v < 8; ++v) acc[dt][v] = 0.f;

  const float* ap = h + (size_t)(row0 + lm) * IN_DIM + koff;
  for (int k0 = 0; k0 < IN_DIM; k0 += 4) {
    v2f a;
    a.x = ap[k0];      // A[M, k0+koff]
    a.y = ap[k0 + 1];  // A[M, k0+koff+1]
    const float* bp = Wn + (size_t)(k0 + koff) * OUT_DIM + col0 + lm;
    #pragma unroll
    for (int dt = 0; dt < 4; ++dt) {
      v2f b;
      b.x = bp[dt * 16];            // B[k0+koff,   n]
      b.y = bp[dt * 16 + OUT_DIM];  // B[k0+koff+1, n]
      acc[dt] = __builtin_amdgcn_wmma_f32_16x16x4_f32(
          false, a, false, b, (short)0, acc[dt], false, false);
    }
  }

  // C layout: VGPR v holds rows v (lanes 0-15) and v+8 (lanes 16-31), N = lane%16
  #pragma unroll
  for (int dt = 0; dt < 4; ++dt)
    #pragma unroll
    for (int v = 0; v < 8; ++v)
      Hn[(size_t)(row0 + v + half * 8) * OUT_DIM + col0 + dt * 16 + lm] =
          acc[dt][v];
}

// ---------------------------------------------------------------------------
// Kernel 3: left/right score vectors: 16384x4 GEMV against u_s / u_n.
// ---------------------------------------------------------------------------
__global__ __launch_bounds__(256) void gat_lr_kernel(
    const float* __restrict__ h, const float* __restrict__ u_s,
    const float* __restrict__ u_n, float* __restrict__ left,
    float* __restrict__ right) {
  const int row = blockIdx.x * blockDim.x + threadIdx.x;  // 0..16383
  float ls[NHEAD] = {0.f, 0.f, 0.f, 0.f};
  float rs[NHEAD] = {0.f, 0.f, 0.f, 0.f};
  const float* hp = h + (size_t)row * IN_DIM;
  for (int k = 0; k < IN_DIM; ++k) {
    const float hv = hp[k];
    #pragma unroll
    for (int hh = 0; hh < NHEAD; ++hh) {
      ls[hh] += hv * u_s[k * NHEAD + hh];
      rs[hh] += hv * u_n[k * NHEAD + hh];
    }
  }
  #pragma unroll
  for (int hh = 0; hh < NHEAD; ++hh) {
    left[(size_t)row * NHEAD + hh]  = ls[hh];
    right[(size_t)row * NHEAD + hh] = rs[hh];
  }
}

// ---------------------------------------------------------------------------
// Kernel 4: rmax[b,h] = max_j right[b,j,h].  Because LeakyReLU is monotonic,
// the softmax row max is LR(l_i + rmax) -> single-pass softmax, no rescaling.
// ---------------------------------------------------------------------------
__global__ __launch_bounds__(256) void gat_rmax_kernel(
    const float* __restrict__ right, float* __restrict__ rmax) {
  const int bh = blockIdx.x;  // b*NHEAD + h
  const int b = bh >> 2, hh = bh & 3;
  __shared__ float red[256];
  float m = -3.402823466e38f;
  for (int j = threadIdx.x; j < SEQ; j += 256)
    m = fmaxf(m, right[((size_t)b * SEQ + j) * NHEAD + hh]);
  red[threadIdx.x] = m;
  __syncthreads();
  for (int s = 128; s > 0; s >>= 1) {
    if (threadIdx.x < s)
      red[threadIdx.x] = fmaxf(red[threadIdx.x], red[threadIdx.x + s]);
    __syncthreads();
  }
  if (threadIdx.x == 0) rmax[bh] = red[0];
}

// ---------------------------------------------------------------------------
// Kernel 5: fused attention: out[b,i,h,:] =
//   ELU( sum_j softmax_j(LR(l_i + r_j)) * Hn[b,j,h,:] )
// Grid: B*H*(SEQ/256) blocks of 128 threads; each wave owns 64 i-rows
// (4 WMMA i-tiles) x full DHEAD=64 (4 WMMA d-tiles) -> 16 v8f accumulators.
// Per 32-j chunk: coalesced right-load + shuffles build P directly in the
// A-matrix layout; 8 K-steps x 16 tiles = 128 f32 WMMAs.
// ---------------------------------------------------------------------------
__global__ __launch_bounds__(128) void gat_attn_kernel(
    const float* __restrict__ Hn, const float* __restrict__ left,
    const float* __restrict__ right, const float* __restrict__ rmax,
    float* __restrict__ out) {
  const int lane  = threadIdx.x & 31;
  const int wave  = threadIdx.x >> 5;
  const int b     = blockIdx.x >> 5;       // 32 blocks per batch
  const int rem   = blockIdx.x & 31;
  const int hh    = rem >> 3;              // head
  const int chunk = rem & 7;               // 256-row chunk within sequence
  const int i_base = chunk * 256 + wave * 64;
  const int half  = lane >> 4;
  const int lm    = lane & 15;
  const int koff  = half * 2;

  const float rmx = rmax[b * NHEAD + hh];

  float lv[4], mv[4], denom[4];
  #pragma unroll
  for (int it = 0; it < 4; ++it) {
    lv[it] = left[((size_t)(b * SEQ + i_base + it * 16 + lm)) * NHEAD + hh];
    float t = lv[it] + rmx;
    mv[it] = (t >= 0.f) ? t : 0.2f * t;   // exact row max of LR(l_i + r_j)
    denom[it] = 0.f;
  }

  v8f acc[4][4];
  #pragma unroll
  for (int it = 0; it < 4; ++it)
    #pragma unroll
    for (int dt = 0; dt < 4; ++dt)
      #pragma unroll
      for (int v = 0; v < 8; ++v) acc[it][dt][v] = 0.f;

  const float* Vbase = Hn + (size_t)(b * SEQ) * OUT_DIM + hh * DHEAD;
  const float* Rbase = right + (size_t)(b * SEQ) * NHEAD + hh;

  for (int j0 = 0; j0 < SEQ; j0 += 32) {
    const float r_c = Rbase[(size_t)(j0 + lane) * NHEAD];  // right[j0+lane]
    #pragma unroll
    for (int kk = 0; kk < 8; ++kk) {
      const int jk = kk * 4 + koff;                 // this half's K columns
      const float r0 = __shfl(r_c, jk, 32);
      const float r1 = __shfl(r_c, jk + 1, 32);

      // V tiles in B-matrix (4x16) layout: rows jk, jk+1 ; N = lane%16
      const float* vp = Vbase + (size_t)(j0 + jk) * OUT_DIM + lm;
      v2f bv[4];
      #pragma unroll
      for (int dt = 0; dt < 4; ++dt) {
        bv[dt].x = vp[dt * 16];
        bv[dt].y = vp[dt * 16 + OUT_DIM];
      }

      #pragma unroll
      for (int it = 0; it < 4; ++it) {
        float t0 = lv[it] + r0; t0 = (t0 >= 0.f) ? t0 : 0.2f * t0;
        float t1 = lv[it] + r1; t1 = (t1 >= 0.f) ? t1 : 0.2f * t1;
        const float p0 = __expf(t0 - mv[it]);
        const float p1 = __expf(t1 - mv[it]);
        denom[it] += p0 + p1;
        v2f a; a.x = p0; a.y = p1;   // A-matrix: M=lm, K columns jk, jk+1
        #pragma unroll
        for (int dt = 0; dt < 4; ++dt) {
          acc[it][dt] = __builtin_amdgcn_wmma_f32_16x16x4_f32(
              false, a, false, bv[dt], (short)0, acc[it][dt], false, false);
        }
      }
    }
  }

  // Lane halves covered complementary column sets {0,1} / {2,3} mod 4
  #pragma unroll
  for (int it = 0; it < 4; ++it) denom[it] += __shfl_xor(denom[it], 16, 32);

  // Epilogue: divide by softmax denominator, ELU, store.
  #pragma unroll
  for (int it = 0; it < 4; ++it) {
    #pragma unroll
    for (int v = 0; v < 8; ++v) {
      const int row = v + half * 8;               // C-row for this VGPR/lane
      const float dv = __shfl(denom[it], v + half * 8, 32);
      const size_t obase =
          ((size_t)(b * SEQ + i_base + it * 16 + row)) * OUT_DIM + hh * DHEAD + lm;
      #pragma unroll
      for (int dt = 0; dt < 4; ++dt) {
        float o = acc[it][dt][v] / dv;
        o = (o > 0.f) ? o : expm1f(o);            // ELU, alpha = 1
        out[obase + dt * 16] = o;
      }
    }
  }
}

// ---------------------------------------------------------------------------
// Launch. Workspace (floats): Hn[16384*256] | left[16384*4] | right[16384*4]
//                             | u_s[256*4] | u_n[256*4] | rmax[32]  (~17.3 MB)
// ---------------------------------------------------------------------------
extern "C" void kernel_launch(void* const* d_in, const int* in_sizes, int n_in,
                              void* d_out, int out_size, void* d_ws,
                              size_t ws_size, hipStream_t stream) {
  const float* h    = (const float*)d_in[0];
  const float* W_s  = (const float*)d_in[1];
  const float* W_n  = (const float*)d_in[2];
  const float* attn = (const float*)d_in[3];
  float* out = (float*)d_out;

  float* ws    = (float*)d_ws;
  float* Hn    = ws;
  float* left  = Hn + (size_t)ROWS * OUT_DIM;
  float* right = left + (size_t)ROWS * NHEAD;
  float* u_s   = right + (size_t)ROWS * NHEAD;
  float* u_n   = u_s + IN_DIM * NHEAD;
  float* rmx   = u_n + IN_DIM * NHEAD;

  gat_prep_kernel<<<1, 256, 0, stream>>>(W_s, W_n, attn, u_s, u_n);
  gat_proj_kernel<<<ROWS / 16, 128, 0, stream>>>(h, W_n, Hn);
  gat_lr_kernel<<<ROWS / 256, 256, 0, stream>>>(h, u_s, u_n, left, right);
  gat_rmax_kernel<<<BATCH * NHEAD, 256, 0, stream>>>(right, rmx);
  gat_attn_kernel<<<BATCH * NHEAD * (SEQ / 256), 128, 0, stream>>>(
      Hn, left, right, rmx, out);
}